// LightGCN_42442866819779
// MI455X (gfx1250) — compile-verified
//
#include <hip/hip_runtime.h>

// ---------------------------------------------------------------------------
// LightGCN forward on MI455X (gfx1250, wave32).
//
// emb' = A_sparse * emb  (COO scatter), 3 layers, mean over {emb0..emb3}.
// Memory/atomic bound: 3.84 GFLOP total vs multi-GB of mostly-L2 traffic.
// Per 16-edge group the message computation is a real matrix product:
//   MSG(16x64) = diag(val)(16x16) x E(16x64)
// computed with V_WMMA_F32_16X16X4_F32 (diag split into four 16x4 slabs,
// fp32 RNE — bit-compatible with scalar FMA accumulation).
//
// Cache policy: embedding tables (38.4 MB) must stay resident in the 192 MB
// L2 because they are randomly re-gathered (~2.56 GB/layer). The edge lists
// (120 MB/layer, streamed once) are loaded NON-TEMPORAL so they do not evict
// the embeddings.
// ---------------------------------------------------------------------------

#define N_USERS   100000
#define N_ITEMS   50000
#define DIM       64
#define NNZ       10000000
#define N_NODES   (N_USERS + N_ITEMS)     // 150000
#define N_LAYERS  3
#define EMB_ELEMS (N_NODES * DIM)         // 9,600,000 floats (38.4 MB)

typedef float v2f __attribute__((ext_vector_type(2)));
typedef float v8f __attribute__((ext_vector_type(8)));
typedef int   v2i __attribute__((ext_vector_type(2)));
typedef int   v4i __attribute__((ext_vector_type(4)));

// ---- init: embA = concat(user,item); acc(d_out) = same --------------------
__global__ void lg_init(const float4* __restrict__ user,
                        const float4* __restrict__ item,
                        float4* __restrict__ embA,
                        float4* __restrict__ acc) {
    int i = blockIdx.x * blockDim.x + threadIdx.x;       // over EMB_ELEMS/4
    const int userV = (N_USERS * DIM) / 4;               // 1,600,000
    float4 v = (i < userV) ? user[i] : item[i - userV];
    embA[i] = v;
    acc[i]  = v;
}

__global__ void lg_zero(float4* __restrict__ p) {
    int i = blockIdx.x * blockDim.x + threadIdx.x;
    p[i] = make_float4(0.f, 0.f, 0.f, 0.f);
}

// acc += e, and zero z (the buffer that becomes the next layer's scatter
// target) in the same pass — saves a full standalone zeroing pass per layer.
__global__ void lg_acc_zero(float4* __restrict__ acc,
                            const float4* __restrict__ e,
                            float4* __restrict__ z) {
    int i = blockIdx.x * blockDim.x + threadIdx.x;
    float4 a = acc[i], b = e[i];
    a.x += b.x; a.y += b.y; a.z += b.z; a.w += b.w;
    acc[i] = a;
    z[i]   = make_float4(0.f, 0.f, 0.f, 0.f);
}

__global__ void lg_scale(float4* __restrict__ p) {
    int i = blockIdx.x * blockDim.x + threadIdx.x;
    float4 a = p[i];
    a.x *= 0.25f; a.y *= 0.25f; a.z *= 0.25f; a.w *= 0.25f;
    p[i] = a;
}

// ---- SpMM: one wave32 handles 16 edges via WMMA ---------------------------
// D(16x16 tile) = sum_k A_k(16x4) x B_k(4x16), A_k = slab of diag(val),
// B_k rows = gathered source-embedding rows (regular temporal loads: these
// hit the L2-resident table). Then 8 scatter-atomics/lane straight out of
// the C/D register layout. Edge-list loads are non-temporal (streamed once).
__global__ void __launch_bounds__(256)
lg_spmm_wmma(const float* __restrict__ emb_in,
             float*       __restrict__ emb_out,
             const float* __restrict__ val,
             const int*   __restrict__ row,
             const int*   __restrict__ col) {
    const int gid  = blockIdx.x * 256 + threadIdx.x;
    const int lane = threadIdx.x & 31;
    const int half = lane >> 4;          // 0: lanes 0-15, 1: lanes 16-31
    const int l    = lane & 15;          // A-matrix row M, B/D column N
    const int e0   = (gid >> 5) << 4;    // first of 16 edges for this wave

    // --- streamed edge data: non-temporal, don't pollute L2 ---------------
    // A-matrix diag value for this lane's row M=l.
    const float myVal = __builtin_nontemporal_load(val + e0 + l);

    // Destination rows for the D layout: VGPR v <-> matrix row v + 8*half.
    // row[e0 + 8*half .. +7]: two aligned NT int4 loads (e0 is 16-aligned).
    v4i r0 = __builtin_nontemporal_load((const v4i*)(row + e0 + 8 * half));
    v4i r1 = __builtin_nontemporal_load((const v4i*)(row + e0 + 8 * half + 4));
    int r[8] = { r0.x, r0.y, r0.z, r0.w, r1.x, r1.y, r1.z, r1.w };

    // Source rows for the B layout: per slab k, this lane supplies
    // B_k[K=2*half][l] and B_k[K=2*half+1][l]  ->  col pairs (4k+2h, 4k+2h+1),
    // which are 8-byte aligned NT int2 loads.
    int cA[4], cB[4];
#pragma unroll
    for (int k = 0; k < 4; ++k) {
        v2i c = __builtin_nontemporal_load((const v2i*)(col + e0 + 4 * k + 2 * half));
        cA[k] = c.x;
        cB[k] = c.y;
    }

    // A-matrix: lane holds A[M=l][K=2*half] (v0) and A[l][2*half+1] (v1);
    // nonzero only where the diagonal crosses this lane's row.
    v2f a[4];
#pragma unroll
    for (int k = 0; k < 4; ++k) {
        a[k].x = (l == 4 * k + 2 * half    ) ? myVal : 0.0f;
        a[k].y = (l == 4 * k + 2 * half + 1) ? myVal : 0.0f;
    }

#pragma unroll
    for (int t = 0; t < 4; ++t) {                 // four 16-wide dim tiles
        v8f d = {};                               // fp32 accumulator tile
#pragma unroll
        for (int k = 0; k < 4; ++k) {
            v2f b;
            b.x = emb_in[cA[k] * DIM + t * 16 + l];   // L2-resident gather
            b.y = emb_in[cB[k] * DIM + t * 16 + l];
            d = __builtin_amdgcn_wmma_f32_16x16x4_f32(
                    /*neg_a=*/false, a[k], /*neg_b=*/false, b,
                    /*c_mod=*/(short)0, d, /*reuse_a=*/false, /*reuse_b=*/false);
        }
#pragma unroll
        for (int v = 0; v < 8; ++v) {
            atomicAdd(&emb_out[r[v] * DIM + t * 16 + l], d[v]);
        }
    }
}

// ---------------------------------------------------------------------------
extern "C" void kernel_launch(void* const* d_in, const int* in_sizes, int n_in,
                              void* d_out, int out_size, void* d_ws, size_t ws_size,
                              hipStream_t stream) {
    const float* user = (const float*)d_in[0];   // [N_USERS, DIM]
    const float* item = (const float*)d_in[1];   // [N_ITEMS, DIM]
    const float* aval = (const float*)d_in[2];   // [NNZ]
    const int*   arow = (const int*)d_in[3];     // [NNZ]
    const int*   acol = (const int*)d_in[4];     // [NNZ]
    float*       out  = (float*)d_out;           // [N_NODES, DIM] mean emb

    float* embA = (float*)d_ws;                  // ping
    float* embB = embA + EMB_ELEMS;              // pong (needs 76.8 MB total)

    const int embBlocks  = (EMB_ELEMS / 4) / 256;        // 9375, exact
    const int spmmBlocks = ((NNZ / 16) * 32) / 256;      // 78125, exact

    lg_init<<<embBlocks, 256, 0, stream>>>((const float4*)user,
                                           (const float4*)item,
                                           (float4*)embA, (float4*)out);
    lg_zero<<<embBlocks, 256, 0, stream>>>((float4*)embB);

    for (int layer = 0; layer < N_LAYERS; ++layer) {
        // scatter embA -> embB (atomics), then out += embB while zeroing embA
        lg_spmm_wmma<<<spmmBlocks, 256, 0, stream>>>(embA, embB, aval, arow, acol);
        lg_acc_zero<<<embBlocks, 256, 0, stream>>>((float4*)out,
                                                   (const float4*)embB,
                                                   (float4*)embA);
        float* tmp = embA; embA = embB; embB = tmp;
    }

    lg_scale<<<embBlocks, 256, 0, stream>>>((float4*)out);
}